// DePatchEfficient_85134841741549
// MI455X (gfx1250) — compile-verified
//
#include <hip/hip_runtime.h>

// ---------------------------------------------------------------------------
// DePatch (overlap-add fold + normalize) for MI455X / gfx1250 — round 2.
//
// x: (B=2, ns=15, nt=15, nu=7, nv=7, pu=4, pv=4, ps=8, pt=8, C=3) fp32
// out: (B=2, U=16, V=16, S=64, T=64, C=3) fp32
// out[b,u,v,s,t,c] = sum over covering patches / overlap_count(u,v,s,t)
//
// Memory-bound gather: 271 MB read once + 25 MB written once ~= 12.7 us
// at 23.3 TB/s. One thread per (b,u,v,s, t-group-of-4):
//   * all 4 sites t=4q..4q+3 share the identical coverage set in every dim
//   * per contributing patch, their 4x3 contributions are 12 contiguous,
//     16B-aligned floats -> three b128 NT loads, accumulators match the
//     output layout exactly (dt-before-C == t-before-C)
//   * overlap count is a power of two -> exact reciprocal via exponent bits
//   * 48 B contiguous NT b128 stores, coalesced across lanes
// ---------------------------------------------------------------------------

typedef float v4f __attribute__((ext_vector_type(4)));

#define C_DIM 3
#define NS 15
#define NT 15
#define NU 7
#define NV 7
#define NPATCH (NS * NT * NU * NV)   // 11025
#define PATCH_ELEMS 1024             // pu*pv*ps*pt = 4*4*8*8

__global__ __launch_bounds__(256)
void depatch_gather4_kernel(const float* __restrict__ x,
                            float* __restrict__ out,
                            int total_groups) {
    int gid = blockIdx.x * blockDim.x + threadIdx.x;
    if (gid >= total_groups) return;

    // gid = (((b*U + u)*V + v)*S + s)*(T/4) + q   with t = 4q + r, r in [0,4)
    int q = gid & 15;
    int s = (gid >> 4) & 63;
    int v = (gid >> 10) & 15;
    int u = (gid >> 14) & 15;
    int b = gid >> 18;

    // Coverage ranges (overlap 1 or 2 per dim), uniform over the t-group.
    // UV: size 4, step 2, n = 7 ; ST: size 8, step 4, n = 15
    int ulo = (u >= 4) ? ((u - 2) >> 1) : 0;
    int uhi = min(NU - 1, u >> 1);
    int vlo = (v >= 4) ? ((v - 2) >> 1) : 0;
    int vhi = min(NV - 1, v >> 1);
    int slo = (s >= 8) ? ((s - 4) >> 2) : 0;
    int shi = min(NS - 1, s >> 2);
    int tlo = max(0, q - 1);
    int thi = min(NT - 1, q);

    const int base_b = b * NPATCH;

    v4f a0 = {0.f, 0.f, 0.f, 0.f};
    v4f a1 = {0.f, 0.f, 0.f, 0.f};
    v4f a2 = {0.f, 0.f, 0.f, 0.f};

    for (int ui = ulo; ui <= uhi; ++ui) {
        const int du = u - 2 * ui;
        for (int vi = vlo; vi <= vhi; ++vi) {
            const int dv = v - 2 * vi;
            const int wuv = (du * 4 + dv) * 64;      // ((du*pv+dv)*ps)*pt
            for (int si = slo; si <= shi; ++si) {
                const int ds = s - 4 * si;
                const int wuvs = wuv + ds * 8;        // + ds*pt
                for (int ti = tlo; ti <= thi; ++ti) {
                    const int dtb = 4 * (q - ti);     // dt of site r=0
                    const int patch = ((si * NT + ti) * NU + ui) * NV + vi;
                    const long idx =
                        ((long)(base_b + patch) * PATCH_ELEMS + (wuvs + dtb)) * C_DIM;
                    // 12 contiguous floats = contributions to 4 sites x 3 ch,
                    // 16B-aligned (idx is a multiple of 12 floats / 48 bytes).
                    const v4f* p = (const v4f*)(x + idx);
                    a0 += __builtin_nontemporal_load(p + 0);
                    a1 += __builtin_nontemporal_load(p + 1);
                    a2 += __builtin_nontemporal_load(p + 2);
                }
            }
        }
    }

    // count = 2^k, identical for all 4 sites -> exact reciprocal.
    int k = (uhi > ulo) + (vhi > vlo) + (shi > slo) + (thi > tlo);
    float inv = __int_as_float((127 - k) << 23);  // 2^-k exactly

    v4f* o = (v4f*)(out + (long)gid * 12);        // 48 B per group, 16B-aligned
    __builtin_nontemporal_store(a0 * inv, o + 0);
    __builtin_nontemporal_store(a1 * inv, o + 1);
    __builtin_nontemporal_store(a2 * inv, o + 2);
}

extern "C" void kernel_launch(void* const* d_in, const int* in_sizes, int n_in,
                              void* d_out, int out_size, void* d_ws, size_t ws_size,
                              hipStream_t stream) {
    const float* x = (const float*)d_in[0];
    float* out = (float*)d_out;

    const int total_groups = out_size / 12;   // B*U*V*S*(T/4) = 524,288
    const int threads = 256;                  // 8 wave32 waves per workgroup
    const int blocks = (total_groups + threads - 1) / threads;

    depatch_gather4_kernel<<<blocks, threads, 0, stream>>>(x, out, total_groups);
}